// Convolution_82678120448582
// MI455X (gfx1250) — compile-verified
//
#include <hip/hip_runtime.h>
#include <stdint.h>

// Problem constants (from reference)
#define HID     768
#define TLEN    3072
#define NBATCH  16
#define KWIN    8
// Tiling
#define TT      48                 // output timesteps per block (mult of 3 and of CHUNK)
#define CHUNK   8                  // rows per async chunk
#define NCHUNK  (TT / CHUNK)       // 6
#define NTHR    (HID / 4)          // 192 threads = 6 waves, 4 channels per thread
#define TILES_T (TLEN / TT)        // 64

#if __has_builtin(__builtin_amdgcn_s_wait_asynccnt)
#define WAIT_ASYNC(n) __builtin_amdgcn_s_wait_asynccnt(n)
#else
#define WAIT_ASYNC(n) asm volatile("s_wait_asynccnt %0" :: "n"(n) : "memory")
#endif

// One 16B row-slice per lane: global -> LDS via the gfx1250 async DMA path.
// Inline asm per CDNA5 ISA (portable across toolchains,
// bypasses the clang builtin's address-space-qualified signature).
__device__ __forceinline__ void async_load_row(const float* gp, float4* lp) {
  uint32_t l = (uint32_t)(uintptr_t)lp;       // low 32 bits = LDS byte offset
  uint64_t g = (uint64_t)(uintptr_t)gp;
  asm volatile("global_load_async_to_lds_b128 %0, %1, off"
               :: "v"(l), "v"(g) : "memory");
}

__global__ __launch_bounds__(NTHR)
void dwconv3_phase_kernel(const float* __restrict__ x,
                          const float* __restrict__ w0, const float* __restrict__ b0,
                          const float* __restrict__ w1, const float* __restrict__ b1,
                          const float* __restrict__ w2, const float* __restrict__ b2,
                          float* __restrict__ out)
{
  __shared__ float4 smem[2][CHUNK][NTHR];     // 48 KB double-buffered staging

  const int tid = threadIdx.x;
  const int b   = blockIdx.x / TILES_T;
  const int t0  = (blockIdx.x % TILES_T) * TT;   // t0 % 3 == 0 by construction
  const int ch  = tid * 4;

  const float* xs = x   + (size_t)b * TLEN * HID + ch;
  float*       os = out + (size_t)b * TLEN * HID + ch;

  // ---- weights: 3 sets x 4 channels x 8 taps, all in registers ----
  float  wreg[3][4][8];
  float4 bias[3];
  {
    const float* wp[3] = {w0, w1, w2};
    const float* bp[3] = {b0, b1, b2};
#pragma unroll
    for (int s = 0; s < 3; ++s) {
#pragma unroll
      for (int c = 0; c < 4; ++c) {
        const float4 lo = *(const float4*)(wp[s] + (size_t)(ch + c) * KWIN);
        const float4 hi = *(const float4*)(wp[s] + (size_t)(ch + c) * KWIN + 4);
        wreg[s][c][0] = lo.x; wreg[s][c][1] = lo.y; wreg[s][c][2] = lo.z; wreg[s][c][3] = lo.w;
        wreg[s][c][4] = hi.x; wreg[s][c][5] = hi.y; wreg[s][c][6] = hi.z; wreg[s][c][7] = hi.w;
      }
      bias[s] = *(const float4*)(bp[s] + ch);
    }
  }

  // ---- history window: win[i] = x row (t0-7+i); zero-pad before t=0 ----
  float4 win[8];
#pragma unroll
  for (int k = 0; k < 7; ++k) {
    const int row = t0 - 7 + k;
    if (row >= 0) win[k] = *(const float4*)(xs + (size_t)row * HID);
    else          win[k] = make_float4(0.f, 0.f, 0.f, 0.f);
  }
  win[7] = make_float4(0.f, 0.f, 0.f, 0.f);

  // ---- prime the async pipeline with chunk 0 ----
#pragma unroll
  for (int r = 0; r < CHUNK; ++r)
    async_load_row(xs + (size_t)(t0 + r) * HID, &smem[0][r][tid]);

#pragma unroll
  for (int c = 0; c < NCHUNK; ++c) {
    const int buf = c & 1;
    if (c + 1 < NCHUNK) {
      // issue next chunk, then wait for current chunk (in-order completion)
#pragma unroll
      for (int r = 0; r < CHUNK; ++r)
        async_load_row(xs + (size_t)(t0 + (c + 1) * CHUNK + r) * HID,
                       &smem[buf ^ 1][r][tid]);
      WAIT_ASYNC(CHUNK);
    } else {
      WAIT_ASYNC(0);
    }
    asm volatile("" ::: "memory");   // keep LDS reads below the wait

#pragma unroll
    for (int r = 0; r < CHUNK; ++r) {
      const int t = t0 + c * CHUNK + r;
      // ring buffer: row (t-7+k) lives in win[(r+k)&7]; new row t -> win[(r+7)&7]
      win[(r + 7) & 7] = smem[buf][r][tid];
      const int s = (c * CHUNK + r) % 3;   // == t % 3 (t0 % 3 == 0); compile-time
      float4 acc = bias[s];
#pragma unroll
      for (int k = 0; k < 8; ++k) {
        const float4 xv = win[(r + k) & 7];
        acc.x += xv.x * wreg[s][0][k];
        acc.y += xv.y * wreg[s][1][k];
        acc.z += xv.z * wreg[s][2][k];
        acc.w += xv.w * wreg[s][3][k];
      }
      *(float4*)(os + (size_t)t * HID) = acc;
    }
  }
}

extern "C" void kernel_launch(void* const* d_in, const int* in_sizes, int n_in,
                              void* d_out, int out_size, void* d_ws, size_t ws_size,
                              hipStream_t stream) {
  (void)in_sizes; (void)n_in; (void)out_size; (void)d_ws; (void)ws_size;
  const float* x  = (const float*)d_in[0];
  const float* rw = (const float*)d_in[1];
  const float* rb = (const float*)d_in[2];
  const float* ow = (const float*)d_in[3];
  const float* ob = (const float*)d_in[4];
  const float* aw = (const float*)d_in[5];
  const float* ab = (const float*)d_in[6];
  float* out = (float*)d_out;

  dim3 grid(NBATCH * TILES_T);   // 16 * 64 = 1024 blocks
  dim3 block(NTHR);              // 192 threads (6 wave32)
  hipLaunchKernelGGL(dwconv3_phase_kernel, grid, block, 0, stream,
                     x, rw, rb, ow, ob, aw, ab, out);
}